// FeatureAttentionLayer_1580547968919
// MI455X (gfx1250) — compile-verified
//
#include <hip/hip_runtime.h>
#include <math.h>

#define B_    16
#define WIN_  128
#define K_    256
#define EMB_  256
#define ALPHA_ 0.2f

typedef __attribute__((ext_vector_type(2))) float v2f;
typedef __attribute__((ext_vector_type(4))) float v4f;
typedef __attribute__((ext_vector_type(8))) float v8f;

// ---------------------------------------------------------------------------
// Kernel A: hi[b,k,e]  = sum_w x[b,w,k] * W1[e,w]
//           hjb[b,k,e] = sum_w x[b,w,k] * W2[e,w] + lin_b[e]
// One wave per 16x16 (k,e) tile; both GEMMs share the A fragment.
// 4096 waves = 1024 blocks x 4 waves. fp32 WMMA 16x16x4.
// ---------------------------------------------------------------------------
__global__ __launch_bounds__(128) void fa_gemm_hi_hj(
    const float* __restrict__ x, const float* __restrict__ lin_w,
    const float* __restrict__ lin_b,
    float* __restrict__ hi, float* __restrict__ hjb)
{
    const int lane = threadIdx.x & 31;
    const int gw   = blockIdx.x * 4 + (threadIdx.x >> 5);   // global wave id 0..4095
    const int b    = gw >> 8;                               // 256 tiles per batch
    const int tt   = gw & 255;
    const int tm   = (tt >> 4) << 4;                        // k-dim tile base
    const int tn   = (tt & 15) << 4;                        // e-dim tile base
    const int r    = lane & 15;
    const int h    = lane >> 4;

    const float* xb = x + (size_t)b * WIN_ * K_;            // x[b][w][k]
    v8f c1 = {};
    v8f c2 = {};
    for (int kk = 0; kk < WIN_; kk += 4) {
        const int w0 = kk + 2 * h;
        // A fragment: A[m=r][K=w0..w0+1]  (xp[b, tm+r, w] = x[b, w, tm+r])
        v2f a;
        a.x = xb[(size_t)w0       * K_ + (tm + r)];
        a.y = xb[(size_t)(w0 + 1) * K_ + (tm + r)];
        // B fragments: B[K=w0..w0+1][n=r] = W{1,2}[tn+r][w0..w0+1]  (contiguous)
        const float* wrow = lin_w + (size_t)(tn + r) * (2 * WIN_);
        v2f b1 = *(const v2f*)(wrow + w0);
        v2f b2 = *(const v2f*)(wrow + WIN_ + w0);
        c1 = __builtin_amdgcn_wmma_f32_16x16x4_f32(false, a, false, b1, (short)0, c1, false, false);
        c2 = __builtin_amdgcn_wmma_f32_16x16x4_f32(false, a, false, b2, (short)0, c2, false, false);
    }

    const float lb = lin_b[tn + r];
    float* hib  = hi  + (size_t)b * K_ * EMB_;
    float* hjbb = hjb + (size_t)b * K_ * EMB_;
#pragma unroll
    for (int v = 0; v < 8; ++v) {
        const int m = tm + v + 8 * h;                       // D layout: m = v + 8h, n = r
        hib [(size_t)m * EMB_ + (tn + r)] = c1[v];
        hjbb[(size_t)m * EMB_ + (tn + r)] = c2[v] + lb;
    }
}

// ---------------------------------------------------------------------------
// Kernel A2: P[b,k] = sum_e a[e]*hi[b,k,e], Q[b,k] = sum_e a[e]*hjb[b,k,e]
// Tiny: 4096 threads, one row pair each.
// ---------------------------------------------------------------------------
__global__ __launch_bounds__(256) void fa_row_dots(
    const float* __restrict__ hi, const float* __restrict__ hjb,
    const float* __restrict__ avec,
    float* __restrict__ P, float* __restrict__ Q)
{
    const int t = blockIdx.x * 256 + threadIdx.x;           // b*K_ + k, 0..4095
    const float* hir = hi  + (size_t)t * EMB_;
    const float* hjr = hjb + (size_t)t * EMB_;
    float p = 0.0f, q = 0.0f;
#pragma unroll 4
    for (int e = 0; e < EMB_; e += 4) {
        const v4f av = *(const v4f*)(avec + e);
        const v4f h1 = *(const v4f*)(hir + e);
        const v4f h2 = *(const v4f*)(hjr + e);
#pragma unroll
        for (int u = 0; u < 4; ++u) {
            p = fmaf(av[u], h1[u], p);
            q = fmaf(av[u], h2[u], q);
        }
    }
    P[t] = p;
    Q[t] = q;
}

// ---------------------------------------------------------------------------
// Kernel B: e[b,i,j] = C1*(P[b,i]+Q[b,j])
//                      + sum_e (C2*a[e]) * |hi[b,i,e]+hjb[b,j,e]|
//                      + bias[i,j]
//           attn[b,i,:] = softmax_j
// leaky(u) = C1*u + C2*|u|, C1=(1+alpha)/2, C2=(1-alpha)/2.
// Hot loop: 2 VALU/elem (|.| is a free src modifier) + b128 LDS broadcasts.
// One 256-thread block per (b,i) row.
// ---------------------------------------------------------------------------
__global__ __launch_bounds__(256) void fa_scores_softmax(
    const float* __restrict__ hi, const float* __restrict__ hjb,
    const float* __restrict__ avec, const float* __restrict__ attn_bias,
    const float* __restrict__ P, const float* __restrict__ Q,
    float* __restrict__ attn)
{
    __shared__ v4f hi_s4[EMB_ / 4];
    __shared__ v4f a2_s4[EMB_ / 4];
    __shared__ float red[256];

    const float C1 = 0.5f * (1.0f + ALPHA_);
    const float C2 = 0.5f * (1.0f - ALPHA_);

    const int j = threadIdx.x;
    const int b = blockIdx.x >> 8;
    const int i = blockIdx.x & 255;

    ((float*)hi_s4)[j] = hi[((size_t)b * K_ + i) * EMB_ + j];
    ((float*)a2_s4)[j] = C2 * avec[j];

    const float* hjrow = hjb + ((size_t)b * K_ + j) * EMB_;
    __builtin_prefetch(hjrow, 0, 0);                        // global_prefetch_b8
    const float pq = P[b * K_ + i] + Q[b * K_ + j];
    __syncthreads();

    float acc = 0.0f;
#pragma unroll 4
    for (int e4 = 0; e4 < EMB_ / 4; ++e4) {
        const v4f hv = *(const v4f*)(hjrow + 4 * e4);
        const v4f hs = hi_s4[e4];                           // ds_load_b128 broadcast
        const v4f as = a2_s4[e4];
#pragma unroll
        for (int u = 0; u < 4; ++u) {
            acc = fmaf(as[u], __builtin_fabsf(hs[u] + hv[u]), acc);
        }
    }
    const float ev = fmaf(C1, pq, acc) + attn_bias[(size_t)i * K_ + j];

    // block-wide max
    red[j] = ev;
    __syncthreads();
    for (int s = 128; s > 0; s >>= 1) {
        if (j < s) red[j] = fmaxf(red[j], red[j + s]);
        __syncthreads();
    }
    const float mx = red[0];
    __syncthreads();

    const float ex = __expf(ev - mx);
    red[j] = ex;
    __syncthreads();
    for (int s = 128; s > 0; s >>= 1) {
        if (j < s) red[j] += red[j + s];
        __syncthreads();
    }
    const float inv = 1.0f / red[0];

    attn[((size_t)b * K_ + i) * K_ + j] = ex * inv;
}

// ---------------------------------------------------------------------------
// Kernel C: out[b,w,i] = sigmoid( sum_j attn[b,i,j] * x[b,w,j] )
// One wave per 16x16 (i,w) tile; 2048 waves = 512 blocks x 4 waves.
// ---------------------------------------------------------------------------
__global__ __launch_bounds__(128) void fa_out_gemm(
    const float* __restrict__ attn, const float* __restrict__ x,
    float* __restrict__ out)
{
    const int lane = threadIdx.x & 31;
    const int gw   = blockIdx.x * 4 + (threadIdx.x >> 5);   // 0..2047
    const int b    = gw >> 7;                               // 128 tiles per batch
    const int tt   = gw & 127;
    const int tm   = (tt >> 3) << 4;                        // i-dim tile base (16 tiles)
    const int tn   = (tt & 7) << 4;                         // w-dim tile base (8 tiles)
    const int r    = lane & 15;
    const int h    = lane >> 4;

    const float* Ab = attn + (size_t)b * K_ * K_;           // attn[b][i][j]
    const float* xb = x    + (size_t)b * WIN_ * K_;         // x[b][w][j] = xp[b][j][w]

    v8f c = {};
    for (int kk = 0; kk < K_; kk += 4) {
        const int k0 = kk + 2 * h;
        v2f a  = *(const v2f*)(Ab + (size_t)(tm + r) * K_ + k0);   // A[m=r][k0..k0+1]
        v2f bb = *(const v2f*)(xb + (size_t)(tn + r) * K_ + k0);   // B[k0..k0+1][n=r]
        c = __builtin_amdgcn_wmma_f32_16x16x4_f32(false, a, false, bb, (short)0, c, false, false);
    }

    float* ob = out + (size_t)b * WIN_ * K_;                // out[b][w][i]
#pragma unroll
    for (int v = 0; v < 8; ++v) {
        const int i = tm + v + 8 * h;
        const int w = tn + r;
        const float s = 1.0f / (1.0f + __expf(-c[v]));
        ob[(size_t)w * K_ + i] = s;
    }
}

// ---------------------------------------------------------------------------
extern "C" void kernel_launch(void* const* d_in, const int* in_sizes, int n_in,
                              void* d_out, int out_size, void* d_ws, size_t ws_size,
                              hipStream_t stream)
{
    const float* x         = (const float*)d_in[0];
    const float* lin_w     = (const float*)d_in[1];
    const float* lin_b     = (const float*)d_in[2];
    const float* a         = (const float*)d_in[3];
    const float* attn_bias = (const float*)d_in[4];
    float* out = (float*)d_out;

    const size_t HKE = (size_t)B_ * K_ * EMB_;              // 1,048,576 floats = 4 MB
    float* hi   = (float*)d_ws;
    float* hjb  = hi + HKE;
    float* attn = hjb + HKE;
    float* P    = attn + HKE;                               // 4096 floats
    float* Q    = P + (size_t)B_ * K_;                      // 4096 floats

    fa_gemm_hi_hj     <<<1024,    128, 0, stream>>>(x, lin_w, lin_b, hi, hjb);
    fa_row_dots       <<<B_,      256, 0, stream>>>(hi, hjb, a, P, Q);
    fa_scores_softmax <<<B_ * K_, 256, 0, stream>>>(hi, hjb, a, attn_bias, P, Q, attn);
    fa_out_gemm       <<<512,     128, 0, stream>>>(attn, x, out);
}